// RandHomoFusedScatterRouter_80427557585601
// MI455X (gfx1250) — compile-verified
//
#include <hip/hip_runtime.h>
#include <hip/hip_bf16.h>
#include <stdint.h>

#define N_TOK   16384
#define DIM     1024
#define PATHS   8
#define CAP     4096
#define RBLK    256                 // tokens per routing block
#define NRBLK   (N_TOK / RBLK)      // 64

// ---- workspace layout (bytes) ----
#define WS_GATE 0                         // float[N_TOK]      (64 KB)
#define WS_PATH 65536                     // int[N_TOK]        (64 KB)
#define WS_HIST 131072                    // int[NRBLK*PATHS]  ( 2 KB)
#define WS_T4S  133120                    // int[PATHS*CAP]    (128 KB)

// -------- CDNA5 async global->LDS helpers (gfx1250) --------
__device__ __forceinline__ void async_ld_b128(const void* g, uint32_t lds_off) {
    // vdst = VGPR holding workgroup-relative LDS byte offset; vaddr = 64-bit global addr
    asm volatile("global_load_async_to_lds_b128 %0, %1, off"
                 :: "v"(lds_off), "v"((uint64_t)(uintptr_t)g)
                 : "memory");
}

__device__ __forceinline__ void wait_async0() {
#if __has_builtin(__builtin_amdgcn_s_wait_asynccnt)
    __builtin_amdgcn_s_wait_asynccnt(0);
    asm volatile("" ::: "memory");
#else
    asm volatile("s_wait_asynccnt 0" ::: "memory");
#endif
}

// -------- Phase 0: init inverse slot map to -1 --------
__global__ void k_init_t4s(int* __restrict__ t4s) {
    int i = blockIdx.x * blockDim.x + threadIdx.x;
    if (i < PATHS * CAP) t4s[i] = -1;
}

// -------- Phase 1: per-token argmax/gate + per-block path histogram --------
__global__ void k_route(const float* __restrict__ scores,
                        float* __restrict__ gate,
                        int*   __restrict__ path,
                        int*   __restrict__ hist) {
    __shared__ int h[PATHS];
    int t = threadIdx.x;
    if (t < PATHS) h[t] = 0;
    __syncthreads();

    int i = blockIdx.x * RBLK + t;
    const float* s = scores + (size_t)i * PATHS;
    float4 a = ((const float4*)s)[0];
    float4 b = ((const float4*)s)[1];
    float v[8] = {a.x, a.y, a.z, a.w, b.x, b.y, b.z, b.w};
    float best = v[0]; int bi = 0;
#pragma unroll
    for (int k = 1; k < 8; ++k)
        if (v[k] > best) { best = v[k]; bi = k; }   // strict '>' == first-max, matches jnp.argmax
    gate[i] = best;
    path[i] = bi;
    atomicAdd(&h[bi], 1);
    __syncthreads();
    if (t < PATHS) hist[blockIdx.x * PATHS + t] = h[t];
}

// -------- Phase 2: exclusive scan of block histograms per path --------
__global__ void k_scan(int* __restrict__ hist) {
    int p = threadIdx.x;
    if (p < PATHS) {
        int run = 0;
        for (int b = 0; b < NRBLK; ++b) {
            int v = hist[b * PATHS + p];
            hist[b * PATHS + p] = run;
            run += v;
        }
    }
}

// -------- Phase 3: in-order intra-block rank -> global slot -> inverse map --------
__global__ void k_slots(const int* __restrict__ path,
                        const int* __restrict__ base,
                        int*       __restrict__ t4s) {
    __shared__ int lp[RBLK];
    int t = threadIdx.x;
    int i = blockIdx.x * RBLK + t;
    int p = path[i];
    lp[t] = p;
    __syncthreads();
    int rank = 0;
    for (int j = 0; j < t; ++j) rank += (lp[j] == p);
    int slot = base[blockIdx.x * PATHS + p] + rank;
    if (slot < CAP) t4s[p * CAP + slot] = i;   // overflow tokens dropped (dump slot)
}

// -------- Phase 4: emit every output row (scaled token or zeros) --------
__global__ void k_emit(const float* __restrict__ x,
                       const float* __restrict__ gate,
                       const int*   __restrict__ t4s,
                       float*       __restrict__ out) {
    __shared__ __attribute__((aligned(16))) float lbuf[DIM];
    int b = blockIdx.x;               // b == p*CAP + slot; out is contiguous in b
    int t = threadIdx.x;              // 0..127
    float* orow = out + (size_t)b * DIM;
    int tok = t4s[b];
    if (tok >= 0) {
        const float* srow = x + (size_t)tok * DIM;
        uint32_t l0 = (uint32_t)(uintptr_t)&lbuf[t * 4];
        uint32_t l1 = (uint32_t)(uintptr_t)&lbuf[512 + t * 4];
        // async Global->LDS (gfx1250 path); each thread loads exactly what it
        // later reads, so per-wave s_wait_asynccnt suffices (no barrier).
        async_ld_b128(srow + t * 4,       l0);
        async_ld_b128(srow + 512 + t * 4, l1);
        float g = gate[tok];
        wait_async0();
        float4 f0 = *(const float4*)&lbuf[t * 4];
        float4 f1 = *(const float4*)&lbuf[512 + t * 4];
        f0.x *= g; f0.y *= g; f0.z *= g; f0.w *= g;
        f1.x *= g; f1.y *= g; f1.z *= g; f1.w *= g;
        *(float4*)(orow + t * 4)       = f0;
        *(float4*)(orow + 512 + t * 4) = f1;
    } else {
        float4 z = make_float4(0.f, 0.f, 0.f, 0.f);
        *(float4*)(orow + t * 4)       = z;
        *(float4*)(orow + 512 + t * 4) = z;
    }
}

extern "C" void kernel_launch(void* const* d_in, const int* in_sizes, int n_in,
                              void* d_out, int out_size, void* d_ws, size_t ws_size,
                              hipStream_t stream) {
    (void)in_sizes; (void)n_in; (void)out_size; (void)ws_size;
    const float* x      = (const float*)d_in[0];
    const float* scores = (const float*)d_in[1];
    float* out = (float*)d_out;

    char* ws = (char*)d_ws;
    float* gate = (float*)(ws + WS_GATE);
    int*   path = (int*)  (ws + WS_PATH);
    int*   hist = (int*)  (ws + WS_HIST);
    int*   t4s  = (int*)  (ws + WS_T4S);

    k_init_t4s<<<(PATHS * CAP + 255) / 256, 256, 0, stream>>>(t4s);
    k_route   <<<NRBLK, RBLK, 0, stream>>>(scores, gate, path, hist);
    k_scan    <<<1, 32, 0, stream>>>(hist);
    k_slots   <<<NRBLK, RBLK, 0, stream>>>(path, hist, t4s);
    k_emit    <<<PATHS * CAP, DIM / 8, 0, stream>>>(x, gate, t4s, out);
}